// ATLASMemory_91147795955944
// MI455X (gfx1250) — compile-verified
//
#include <hip/hip_runtime.h>

typedef float v2f __attribute__((ext_vector_type(2)));
typedef float v8f __attribute__((ext_vector_type(8)));

namespace {
constexpr int Bsz = 4;
constexpr int L   = 8192;
constexpr int D   = 512;
constexpr int W   = 64;
constexpr int BW  = Bsz * W;   // 256 token rows per window
constexpr int NW  = L / W;     // 128 windows
}

// element offset of token row r (r in [0,256)) of window t inside a (B,L,D) tensor
__device__ __forceinline__ size_t tok_off(int r, int t) {
  return ((size_t)(r >> 6) * L + (size_t)t * W + (size_t)(r & 63)) * D;
}

__device__ __forceinline__ v8f wmma4(v2f a, v2f b, v8f c) {
  // V_WMMA_F32_16X16X4_F32 : D = A(16x4) * B(4x16) + C, fp32 throughout
  return __builtin_amdgcn_wmma_f32_16x16x4_f32(false, a, false, b, (short)0, c,
                                               false, false);
}

// ---------------------------------------------------------------------------
// Kernel 1:  T[e,r] = g_r * ( V[r,e] - sum_d M[e,d]*K[r,d] )        (512 x 256)
// GEMM: A = M (row-major, ld=512), B = K^T (loaded via per-token row pointers)
// ---------------------------------------------------------------------------
__global__ __launch_bounds__(128) void atlas_k1(
    const float* __restrict__ Mprev, const float* __restrict__ keys,
    const float* __restrict__ values, const float* __restrict__ gammas,
    float* __restrict__ T, int t) {
  const int lane = threadIdx.x & 31;
  const int wave = threadIdx.x >> 5;
  const int lrow = lane & 15;
  const int lhi  = lane >> 4;
  const int row0 = blockIdx.y * 64 + wave * 16;  // e (output row of M-dim)
  const int col0 = blockIdx.x * 64;              // r (token column)

  v8f acc[4];
#pragma unroll
  for (int f = 0; f < 4; ++f)
#pragma unroll
    for (int i = 0; i < 8; ++i) acc[f][i] = 0.0f;

  const float* arow = Mprev + (size_t)(row0 + lrow) * D;
  const float* brow[4];
#pragma unroll
  for (int f = 0; f < 4; ++f)
    brow[f] = keys + tok_off(col0 + f * 16 + lrow, t);

  for (int k0 = 0; k0 < D; k0 += 4) {
    const int kb = k0 + lhi * 2;
    const v2f a = *(const v2f*)(arow + kb);
#pragma unroll
    for (int f = 0; f < 4; ++f) {
      const v2f b = *(const v2f*)(brow[f] + kb);  // B[k,r] = K[r,k]
      acc[f] = wmma4(a, b, acc[f]);
    }
  }

#pragma unroll
  for (int f = 0; f < 4; ++f) {
    const int r = col0 + f * 16 + lrow;
    const size_t voff = tok_off(r, t);
    const float g = gammas[(size_t)(r >> 6) * L + (size_t)t * W + (r & 63)];
#pragma unroll
    for (int v = 0; v < 8; ++v) {
      const int e = row0 + lhi * 8 + v;
      T[(size_t)e * BW + r] = g * (values[voff + e] - acc[f][v]);
    }
  }
}

// ---------------------------------------------------------------------------
// Kernel 2:  M_new[e,j] = a * M[e,j] + sum_r T[e,r] * K[r,j]        (512 x 512)
// GEMM: A = T (row-major, ld=256), B = K (row-major via token row pointers)
// ---------------------------------------------------------------------------
__global__ __launch_bounds__(128) void atlas_k2(
    const float* __restrict__ Mprev, const float* __restrict__ T,
    const float* __restrict__ keys, const float* __restrict__ alpha_p,
    float* __restrict__ Mnew, int t) {
  const int lane = threadIdx.x & 31;
  const int wave = threadIdx.x >> 5;
  const int lrow = lane & 15;
  const int lhi  = lane >> 4;
  const int row0 = blockIdx.y * 64 + wave * 16;  // e
  const int col0 = blockIdx.x * 64;              // j

  v8f acc[4];
#pragma unroll
  for (int f = 0; f < 4; ++f)
#pragma unroll
    for (int i = 0; i < 8; ++i) acc[f][i] = 0.0f;

  const float* arow = T + (size_t)(row0 + lrow) * BW;

  for (int k0 = 0; k0 < BW; k0 += 4) {
    const int kb = k0 + lhi * 2;  // even, kb and kb+1 in same 64-token batch
    const v2f a = *(const v2f*)(arow + kb);
    const float* b0 = keys + tok_off(kb, t);
    const float* b1 = keys + tok_off(kb + 1, t);
#pragma unroll
    for (int f = 0; f < 4; ++f) {
      const int n = col0 + f * 16 + lrow;
      v2f b;
      b.x = b0[n];
      b.y = b1[n];
      acc[f] = wmma4(a, b, acc[f]);
    }
  }

  const float a_sig = 1.0f / (1.0f + __expf(-alpha_p[0]));
#pragma unroll
  for (int f = 0; f < 4; ++f) {
    const int n = col0 + f * 16 + lrow;
#pragma unroll
    for (int v = 0; v < 8; ++v) {
      const int e = row0 + lhi * 8 + v;
      Mnew[(size_t)e * D + n] = a_sig * Mprev[(size_t)e * D + n] + acc[f][v];
    }
  }
}

// ---------------------------------------------------------------------------
// Kernel 3:  out[r,e] = sum_d Q[r,d] * M_new[e,d]                   (256 x 512)
// GEMM: A = Q (token rows), B = M_new^T (per-output-feature row pointers)
// ---------------------------------------------------------------------------
__global__ __launch_bounds__(128) void atlas_k3(
    const float* __restrict__ Mnew, const float* __restrict__ queries,
    float* __restrict__ out, int t) {
  const int lane = threadIdx.x & 31;
  const int wave = threadIdx.x >> 5;
  const int lrow = lane & 15;
  const int lhi  = lane >> 4;
  const int row0 = blockIdx.y * 64 + wave * 16;  // r (token)
  const int col0 = blockIdx.x * 64;              // e

  v8f acc[4];
#pragma unroll
  for (int f = 0; f < 4; ++f)
#pragma unroll
    for (int i = 0; i < 8; ++i) acc[f][i] = 0.0f;

  const float* arow = queries + tok_off(row0 + lrow, t);
  const float* brow[4];
#pragma unroll
  for (int f = 0; f < 4; ++f)
    brow[f] = Mnew + (size_t)(col0 + f * 16 + lrow) * D;  // B[k,e] = M[e,k]

  for (int k0 = 0; k0 < D; k0 += 4) {
    const int kb = k0 + lhi * 2;
    const v2f a = *(const v2f*)(arow + kb);
#pragma unroll
    for (int f = 0; f < 4; ++f) {
      const v2f b = *(const v2f*)(brow[f] + kb);
      acc[f] = wmma4(a, b, acc[f]);
    }
  }

#pragma unroll
  for (int f = 0; f < 4; ++f) {
    const int n = col0 + f * 16 + lrow;
#pragma unroll
    for (int v = 0; v < 8; ++v) {
      const int r = row0 + lhi * 8 + v;
      out[tok_off(r, t) + n] = acc[f][v];
    }
  }
}

// ---------------------------------------------------------------------------
extern "C" void kernel_launch(void* const* d_in, const int* in_sizes, int n_in,
                              void* d_out, int out_size, void* d_ws,
                              size_t ws_size, hipStream_t stream) {
  const float* keys    = (const float*)d_in[0];
  const float* values  = (const float*)d_in[1];
  const float* queries = (const float*)d_in[2];
  const float* gammas  = (const float*)d_in[3];
  const float* alpha   = (const float*)d_in[4];
  const float* M0      = (const float*)d_in[5];
  float* out = (float*)d_out;

  // workspace: double-buffered M (2 * 512*512) + T (512*256) = 2.5 MB
  float* Mb0 = (float*)d_ws;
  float* Mb1 = Mb0 + (size_t)D * D;
  float* T   = Mb1 + (size_t)D * D;

  const dim3 blk(128);
  const dim3 g1(BW / 64, D / 64);  // (4, 8)  -> T      (512 x 256)
  const dim3 g2(D / 64, D / 64);   // (8, 8)  -> M_new  (512 x 512)
  const dim3 g3(D / 64, BW / 64);  // (8, 4)  -> out    (256 x 512)

  for (int t = 0; t < NW; ++t) {
    const float* Mprev = (t == 0) ? M0 : ((t & 1) ? Mb0 : Mb1);
    float* Mnew = (t & 1) ? Mb1 : Mb0;
    atlas_k1<<<g1, blk, 0, stream>>>(Mprev, keys, values, gammas, T, t);
    atlas_k2<<<g2, blk, 0, stream>>>(Mprev, T, keys, alpha, Mnew, t);
    atlas_k3<<<g3, blk, 0, stream>>>(Mnew, queries, out, t);
  }
}